// FFTSelector_67826123538942
// MI455X (gfx1250) — compile-verified
//
#include <hip/hip_runtime.h>
#include <hip/hip_bf16.h>

#define BATCH 4
#define TT    288
#define NN    207
#define DD    152
#define ND    31464          // NN*DD
#define ND4   7866           // ND/4, exact
#define FFDIM 256
#define TOPK  6

typedef __attribute__((ext_vector_type(2))) float v2f;
typedef __attribute__((ext_vector_type(4))) float v4f;
typedef __attribute__((ext_vector_type(8))) float v8f;

// ---- workspace layout (float offsets) ----
#define WS_WQ   0                        // ND   : column sums of Wq
#define WS_WK   (WS_WQ + ND)             // ND   : column sums of Wk
#define WS_BS   (WS_WK + ND)             // 2    : bias sums (q,k)
#define WS_QS   (WS_BS + 4)              // BATCH*TT : Qs[b][t]
#define WS_KS   (WS_QS + BATCH*TT)       // BATCH*TT : Ks[b][t]
#define WS_CORR (WS_KS + BATCH*TT)       // TT*TT
#define WS_IDX  (WS_CORR + TT*TT)        // TT*TOPK ints

__device__ __forceinline__ float neg_inf() { return __int_as_float(0xff800000); }

// --- 1: column sums of Wq, Wk over FF rows (coalesced per f) ---
__global__ void colsum_kernel(const float* __restrict__ Wq,
                              const float* __restrict__ Wk,
                              float* __restrict__ ws) {
    int j = blockIdx.x * blockDim.x + threadIdx.x;
    if (j >= ND) return;
    float sq = 0.f, sk = 0.f;
    #pragma unroll 4
    for (int f = 0; f < FFDIM; ++f) {
        sq += Wq[(size_t)f * ND + j];
        sk += Wk[(size_t)f * ND + j];
    }
    ws[WS_WQ + j] = sq;
    ws[WS_WK + j] = sk;
}

// --- 2: bias sums (FF == 256 == blockDim) ---
__global__ void biassum_kernel(const float* __restrict__ bq,
                               const float* __restrict__ bk,
                               float* __restrict__ ws) {
    __shared__ float sq[256], sk[256];
    int t = threadIdx.x;
    sq[t] = bq[t]; sk[t] = bk[t];
    __syncthreads();
    for (int s = 128; s > 0; s >>= 1) {
        if (t < s) { sq[t] += sq[t + s]; sk[t] += sk[t + s]; }
        __syncthreads();
    }
    if (t == 0) { ws[WS_BS] = sq[0]; ws[WS_BS + 1] = sk[0]; }
}

// --- 3: Qs/Ks: one block per (b,t) row; dual dot product vs colsum vectors ---
__global__ void rowdot_kernel(const float* __restrict__ X, float* __restrict__ ws) {
    int r = blockIdx.x;  // r = b*TT + t
    const v4f* xr = (const v4f*)(X + (size_t)r * ND);
    const v4f* wq = (const v4f*)(ws + WS_WQ);
    const v4f* wk = (const v4f*)(ws + WS_WK);
    float pq = 0.f, pk = 0.f;
    for (int i = threadIdx.x; i < ND4; i += blockDim.x) {
        v4f xv = xr[i];
        v4f q  = wq[i];
        v4f k  = wk[i];
        pq += xv.x*q.x + xv.y*q.y + xv.z*q.z + xv.w*q.w;
        pk += xv.x*k.x + xv.y*k.y + xv.z*k.z + xv.w*k.w;
    }
    __shared__ float rq[256], rk[256];
    int t = threadIdx.x;
    rq[t] = pq; rk[t] = pk;
    __syncthreads();
    for (int s = 128; s > 0; s >>= 1) {
        if (t < s) { rq[t] += rq[t + s]; rk[t] += rk[t + s]; }
        __syncthreads();
    }
    if (t == 0) {
        ws[WS_QS + r] = rq[0] + ws[WS_BS];
        ws[WS_KS + r] = rk[0] + ws[WS_BS + 1];
    }
}

// --- 4: corr = (1/(129*B)) * Qs^T(288x4) x Ks(4x288) via V_WMMA_F32_16X16X4_F32 ---
// One wave (32 threads) per 16x16 output tile; 18x18 tiles exactly cover 288x288.
__global__ void __launch_bounds__(32) corr_wmma_kernel(float* __restrict__ ws) {
    const float* qs = ws + WS_QS;   // [b][t]
    const float* ks = ws + WS_KS;   // [b][t]
    float* corr = ws + WS_CORR;

    const int ti   = blockIdx.x;    // row tile
    const int tj   = blockIdx.y;    // col tile
    const int lane = threadIdx.x;   // 0..31
    const int half = lane >> 4;     // 0 or 1
    const int mn   = lane & 15;
    const int kb   = half * 2;      // K pair base: {0,1} or {2,3}
    const int m_g  = ti * 16 + mn;
    const int n_g  = tj * 16 + mn;

    // A = Qs^T (16x4): lane holds row M=mn; VGPR0=K(kb), VGPR1=K(kb+1)
    v2f a, b;
    a.x = qs[(kb + 0) * TT + m_g];
    a.y = qs[(kb + 1) * TT + m_g];
    // B = Ks (4x16): lane holds col N=mn; VGPR0=K(kb), VGPR1=K(kb+1)
    b.x = ks[(kb + 0) * TT + n_g];
    b.y = ks[(kb + 1) * TT + n_g];

    v8f c = {};
    c = __builtin_amdgcn_wmma_f32_16x16x4_f32(
            /*neg_a=*/false, a, /*neg_b=*/false, b,
            /*c_mod=*/(short)0, c, /*reuse_a=*/false, /*reuse_b=*/false);

    const float scale = 1.0f / (129.0f * (float)BATCH);
    #pragma unroll
    for (int v = 0; v < 8; ++v) {
        int m = ti * 16 + v + 8 * half;   // D layout: VGPR v -> M = v + 8*(lane/16)
        int n = tj * 16 + mn;             //           lane%16 -> N
        corr[m * TT + n] = (m == n) ? neg_inf() : c[v] * scale;
    }
}

// --- 5: per-row top-6 (descending, ties -> smaller index), then sort by index ---
__global__ void topk_kernel(float* __restrict__ ws, float* __restrict__ out) {
    int t = blockIdx.x * blockDim.x + threadIdx.x;
    if (t >= TT) return;
    const float* row = ws + WS_CORR + (size_t)t * TT;
    float bv[TOPK];
    int   bi[TOPK];
    #pragma unroll
    for (int i = 0; i < TOPK; ++i) { bv[i] = neg_inf(); bi[i] = 0x7fffffff; }
    for (int n = 0; n < TT; ++n) {
        float v = row[n];
        if (v > bv[TOPK - 1]) {                 // strict >: earlier index wins ties
            int p = TOPK - 1;
            while (p > 0 && v > bv[p - 1]) {
                bv[p] = bv[p - 1]; bi[p] = bi[p - 1]; --p;
            }
            bv[p] = v; bi[p] = n;
        }
    }
    // sort the 6 hits by index ascending (indices are distinct)
    #pragma unroll
    for (int i = 1; i < TOPK; ++i) {
        float v = bv[i]; int id = bi[i];
        int p = i;
        while (p > 0 && bi[p - 1] > id) {
            bv[p] = bv[p - 1]; bi[p] = bi[p - 1]; --p;
        }
        bv[p] = v; bi[p] = id;
    }
    int* idxws = (int*)(ws + WS_IDX);
    #pragma unroll
    for (int k = 0; k < TOPK; ++k) {
        out[t * TOPK + k]             = bv[k];           // sorted_values
        out[TT * TOPK + t * TOPK + k] = (float)bi[k];    // sorted_indices
        idxws[t * TOPK + k]           = bi[k];
    }
}

// --- 6: out3[b,t,k,:,:] = X[b, idx[t,k], :, :]  (dominant 870 MB stream) ---
// Loads use default (cacheable) policy so X (145 MB) stays hot in the 192 MB L2;
// stores are non-temporal (TH=NT) so the write-once output stream does not
// evict X from L2 (HBM traffic ~1.0 GB instead of ~1.9 GB).
__global__ void gather_kernel(const float* __restrict__ X,
                              const float* __restrict__ ws,
                              float* __restrict__ out) {
    const int* idxws = (const int*)(ws + WS_IDX);
    int bx = blockIdx.x;                 // ((b*TT + t)*TOPK + k)
    int k  = bx % TOPK;
    int t  = (bx / TOPK) % TT;
    int b  = bx / (TOPK * TT);
    int src_t = idxws[t * TOPK + k];
    const v4f* s = (const v4f*)(X + ((size_t)b * TT + src_t) * ND);
    v4f*       d = (v4f*)(out + 2 * TT * TOPK + (size_t)bx * ND);
    for (int i = threadIdx.x; i < ND4; i += blockDim.x) {
        v4f v = s[i];
        __builtin_nontemporal_store(v, d + i);
    }
}

extern "C" void kernel_launch(void* const* d_in, const int* in_sizes, int n_in,
                              void* d_out, int out_size, void* d_ws, size_t ws_size,
                              hipStream_t stream) {
    const float* X  = (const float*)d_in[0];
    const float* Wq = (const float*)d_in[1];
    const float* bq = (const float*)d_in[2];
    const float* Wk = (const float*)d_in[3];
    const float* bk = (const float*)d_in[4];
    float* out = (float*)d_out;
    float* ws  = (float*)d_ws;

    colsum_kernel <<<(ND + 255) / 256, 256, 0, stream>>>(Wq, Wk, ws);
    biassum_kernel<<<1, 256, 0, stream>>>(bq, bk, ws);
    rowdot_kernel <<<BATCH * TT, 256, 0, stream>>>(X, ws);
    dim3 gtiles(TT / 16, TT / 16);       // 18 x 18
    corr_wmma_kernel<<<gtiles, 32, 0, stream>>>(ws);
    topk_kernel   <<<(TT + 63) / 64, 64, 0, stream>>>(ws, out);
    gather_kernel <<<BATCH * TT * TOPK, 256, 0, stream>>>(X, ws, out);
}